// PythonNeuralODECell_10247791968650
// MI455X (gfx1250) — compile-verified
//
#include <hip/hip_runtime.h>
#include <math.h>

typedef __attribute__((ext_vector_type(2))) float v2f;
typedef __attribute__((ext_vector_type(8))) float v8f;

#define T_DIM  1024
#define B_DIM  32
#define D_DIM  768
#define N_DIM  64
#define PROJ_C 256   // 4*n : [k | v | q | m]

// ---------------------------------------------------------------------------
// Kernel A: proj[row, c] = sum_d x[row, d] * W[c, d]
// fp32 WMMA 16x16x4. One wave -> 16x64 output tile (4 chained accumulators).
// ---------------------------------------------------------------------------
__global__ __launch_bounds__(256) void proj_gemm_wmma(
    const float* __restrict__ x, const float* __restrict__ W,
    float* __restrict__ proj)
{
    const int lane  = threadIdx.x & 31;
    const int wave  = blockIdx.x * (blockDim.x >> 5) + (threadIdx.x >> 5);
    const int mtile = wave >> 2;        // 0..2047
    const int nquad = wave & 3;         // 0..3  -> 64-col slab
    const int mbase = mtile * 16;
    const int nbase = nquad * 64;

    const int mrow  = mbase + (lane & 15);       // A: M = lane%16
    const int khalf = (lane >> 4) * 2;           // A/B: K pair select
    const int col0  = nbase + (lane & 15);       // B/C: N = lane%16

    const float* __restrict__ xrow = x + (size_t)mrow * D_DIM;
    const float* __restrict__ w0 = W + (size_t)(col0 +  0) * D_DIM;
    const float* __restrict__ w1 = W + (size_t)(col0 + 16) * D_DIM;
    const float* __restrict__ w2 = W + (size_t)(col0 + 32) * D_DIM;
    const float* __restrict__ w3 = W + (size_t)(col0 + 48) * D_DIM;

    v8f acc0 = {}, acc1 = {}, acc2 = {}, acc3 = {};

    for (int kk = 0; kk < D_DIM; kk += 4) {
        const int ka = kk + khalf;
        v2f a;  a.x  = xrow[ka]; a.y  = xrow[ka + 1];
        v2f b0; b0.x = w0[ka];   b0.y = w0[ka + 1];
        v2f b1; b1.x = w1[ka];   b1.y = w1[ka + 1];
        v2f b2; b2.x = w2[ka];   b2.y = w2[ka + 1];
        v2f b3; b3.x = w3[ka];   b3.y = w3[ka + 1];
        acc0 = __builtin_amdgcn_wmma_f32_16x16x4_f32(false, a, false, b0, (short)0, acc0, false, false);
        acc1 = __builtin_amdgcn_wmma_f32_16x16x4_f32(false, a, false, b1, (short)0, acc1, false, false);
        acc2 = __builtin_amdgcn_wmma_f32_16x16x4_f32(false, a, false, b2, (short)0, acc2, false, false);
        acc3 = __builtin_amdgcn_wmma_f32_16x16x4_f32(false, a, false, b3, (short)0, acc3, false, false);
    }

    const int rbase = mbase + 8 * (lane >> 4);   // C: row = vgpr + 8*(lane>=16)
#pragma unroll
    for (int r = 0; r < 8; ++r) {
        const size_t row = (size_t)(rbase + r) * PROJ_C;
        proj[row + col0 +  0] = acc0[r];
        proj[row + col0 + 16] = acc1[r];
        proj[row + col0 + 32] = acc2[r];
        proj[row + col0 + 48] = acc3[r];
    }
}

// ---------------------------------------------------------------------------
// Kernel B: normalize k (cols 0..63) and m (cols 192..255) per row.
// ---------------------------------------------------------------------------
__global__ __launch_bounds__(64) void normalize_km(float* __restrict__ proj)
{
    const int row = blockIdx.x;
    const int tid = threadIdx.x;          // 0..63
    float* p = proj + (size_t)row * PROJ_C;

    float kvl = p[tid];
    float mvl = p[192 + tid];
    float pk = kvl * kvl, pm = mvl * mvl;
#pragma unroll
    for (int off = 16; off > 0; off >>= 1) {
        pk += __shfl_xor(pk, off);
        pm += __shfl_xor(pm, off);
    }
    __shared__ float sk[2], sm[2];
    if ((tid & 31) == 0) { sk[tid >> 5] = pk; sm[tid >> 5] = pm; }
    __syncthreads();
    const float nk = sqrtf(sk[0] + sk[1]) + 1e-6f;
    const float nm = sqrtf(sm[0] + sm[1]) + 1e-6f;
    p[tid]       = kvl / nk;
    p[192 + tid] = mvl / nm;
}

// ---------------------------------------------------------------------------
// Kernel C: sequential ODE scan. One block (256 threads) per batch b.
// Thread tid owns row i = tid>>2, cols [j0, j0+16) of BOTH S and G in regs.
// After the 4-lane xor-butterfly, ALL lanes of a row group hold the full dot
// products -> deltas/gates computed redundantly per lane: NO barriers, NO LDS
// inside the RK4 chain. Barriers: 2 per timestep (around the row load).
// ---------------------------------------------------------------------------
__device__ __forceinline__ void ode_eval_regs(
    const float Xc[16], const float Yc[16],
    const float kr[16], const float mr[16], const float vi,
    float& dSi, float& dGi, float& gS, float& gG)
{
    float p_sk = 0.f, p_gm = 0.f, p_gk = 0.f, p_sm = 0.f;
#pragma unroll
    for (int r = 0; r < 16; ++r) {
        p_sk += Xc[r] * kr[r];  p_sm += Xc[r] * mr[r];
        p_gm += Yc[r] * mr[r];  p_gk += Yc[r] * kr[r];
    }
    // full reduction across the 4 lanes owning this row; result in all 4 lanes
    p_sk += __shfl_xor(p_sk, 1); p_sk += __shfl_xor(p_sk, 2);
    p_gm += __shfl_xor(p_gm, 1); p_gm += __shfl_xor(p_gm, 2);
    p_gk += __shfl_xor(p_gk, 1); p_gk += __shfl_xor(p_gk, 2);
    p_sm += __shfl_xor(p_sm, 1); p_sm += __shfl_xor(p_sm, 2);
    dSi = vi - p_sk;                       // delta_S[i]
    dGi = dSi - p_gm;                      // delta_G[i]
    gS  = 1.0f / (1.0f + __expf(-p_gk));   // gate_S[i] = sigmoid((G.k)[i])
    gG  = 1.0f / (1.0f + __expf(-p_sm));   // gate_G[i] = sigmoid((S.m)[i])
}

__global__ __launch_bounds__(256) void scan_kernel(
    const float* __restrict__ proj,
    const float* __restrict__ S0, const float* __restrict__ G0,
    const int* __restrict__ nsteps_p,
    float* __restrict__ out, float* __restrict__ S_out, float* __restrict__ G_out)
{
    const int b   = blockIdx.x;
    const int tid = threadIdx.x;
    const int i   = tid >> 2;
    const int seg = tid & 3;
    const int j0  = seg * 16;
    const int n_steps = nsteps_p[0];
    const float dt  = 1.0f / (float)n_steps;
    const float hdt = 0.5f * dt;
    const float w6  = dt * (1.0f / 6.0f);

    float Sb[16], Gb[16];
    {
        const float* s0 = S0 + ((size_t)b * N_DIM + i) * N_DIM + j0;
        const float* g0 = G0 + ((size_t)b * N_DIM + i) * N_DIM + j0;
#pragma unroll
        for (int r = 0; r < 16; ++r) { Sb[r] = s0[r]; Gb[r] = g0[r]; }
    }

    __shared__ float kv[64], vv[64], qv[64], mv[64];

    // software-pipelined row fetch
    float val = proj[(size_t)b * PROJ_C + tid];

    for (int t = 0; t < T_DIM; ++t) {
        __syncthreads();                       // previous readers done
        if      (tid <  64) kv[tid]       = val;
        else if (tid < 128) vv[tid - 64]  = val;
        else if (tid < 192) qv[tid - 128] = val;
        else                mv[tid - 192] = val;
        __syncthreads();

        // hoist timestep-invariant vector segments into registers
        float kr[16], mr[16];
#pragma unroll
        for (int r = 0; r < 16; ++r) { kr[r] = kv[j0 + r]; mr[r] = mv[j0 + r]; }
        const float vi = vv[i];

        // prefetch next timestep's row while we crunch
        float vnext = 0.f;
        if (t + 1 < T_DIM)
            vnext = proj[((size_t)(t + 1) * B_DIM + b) * PROJ_C + tid];

        for (int sub = 0; sub < n_steps; ++sub) {
            float accS[16], accG[16], Sc[16], Gc[16];
            float dSi, dGi, gS, gG;

            // stage 1 at (Sb, Gb)
            ode_eval_regs(Sb, Gb, kr, mr, vi, dSi, dGi, gS, gG);
#pragma unroll
            for (int r = 0; r < 16; ++r) {
                const float d = (gS - 1.0f) * Sb[r] + dSi * kr[r];
                const float e = (gG - 1.0f) * Gb[r] + dGi * mr[r];
                accS[r] = d;                  accG[r] = e;
                Sc[r] = Sb[r] + hdt * d;      Gc[r] = Gb[r] + hdt * e;
            }
            // stage 2
            ode_eval_regs(Sc, Gc, kr, mr, vi, dSi, dGi, gS, gG);
#pragma unroll
            for (int r = 0; r < 16; ++r) {
                const float d = (gS - 1.0f) * Sc[r] + dSi * kr[r];
                const float e = (gG - 1.0f) * Gc[r] + dGi * mr[r];
                accS[r] += 2.0f * d;          accG[r] += 2.0f * e;
                Sc[r] = Sb[r] + hdt * d;      Gc[r] = Gb[r] + hdt * e;
            }
            // stage 3
            ode_eval_regs(Sc, Gc, kr, mr, vi, dSi, dGi, gS, gG);
#pragma unroll
            for (int r = 0; r < 16; ++r) {
                const float d = (gS - 1.0f) * Sc[r] + dSi * kr[r];
                const float e = (gG - 1.0f) * Gc[r] + dGi * mr[r];
                accS[r] += 2.0f * d;          accG[r] += 2.0f * e;
                Sc[r] = Sb[r] + dt * d;       Gc[r] = Gb[r] + dt * e;
            }
            // stage 4 + combine
            ode_eval_regs(Sc, Gc, kr, mr, vi, dSi, dGi, gS, gG);
#pragma unroll
            for (int r = 0; r < 16; ++r) {
                const float d = (gS - 1.0f) * Sc[r] + dSi * kr[r];
                const float e = (gG - 1.0f) * Gc[r] + dGi * mr[r];
                Sb[r] += w6 * (accS[r] + d);
                Gb[r] += w6 * (accG[r] + e);
            }
        }

        // output: Sq = S.q ; out = Sq * silu(Sq) = Sq^2 * sigmoid(Sq)
        float pq = 0.f;
#pragma unroll
        for (int r = 0; r < 16; ++r) pq += Sb[r] * qv[j0 + r];
        pq += __shfl_xor(pq, 1); pq += __shfl_xor(pq, 2);
        if (seg == 0) {
            const float sg = 1.0f / (1.0f + __expf(-pq));
            out[((size_t)t * B_DIM + b) * N_DIM + i] = pq * pq * sg;
        }

        val = vnext;
    }

    // final states
    float* so = S_out + ((size_t)b * N_DIM + i) * N_DIM + j0;
    float* go = G_out + ((size_t)b * N_DIM + i) * N_DIM + j0;
#pragma unroll
    for (int r = 0; r < 16; ++r) { so[r] = Sb[r]; go[r] = Gb[r]; }
}

// ---------------------------------------------------------------------------
extern "C" void kernel_launch(void* const* d_in, const int* in_sizes, int n_in,
                              void* d_out, int out_size, void* d_ws, size_t ws_size,
                              hipStream_t stream)
{
    const float* x      = (const float*)d_in[0];  // [T,B,dim]
    const float* S0     = (const float*)d_in[1];  // [B,n,n]
    const float* G0     = (const float*)d_in[2];  // [B,n,n]
    const float* W      = (const float*)d_in[3];  // [4n,dim]
    const int*   nsteps = (const int*)  d_in[4];  // scalar

    float* proj  = (float*)d_ws;                  // [T*B, 256] = 32 MB scratch
    float* out   = (float*)d_out;                 // [T,B,n]
    float* S_out = out + (size_t)T_DIM * B_DIM * N_DIM;
    float* G_out = S_out + (size_t)B_DIM * N_DIM * N_DIM;

    proj_gemm_wmma<<<1024, 256, 0, stream>>>(x, W, proj);
    normalize_km<<<T_DIM * B_DIM, 64, 0, stream>>>(proj);
    scan_kernel<<<B_DIM, 256, 0, stream>>>(proj, S0, G0, nsteps, out, S_out, G_out);
}